// SpatioTemporalRegressor_15281493639570
// MI455X (gfx1250) — compile-verified
//
#include <hip/hip_runtime.h>
#include <hip/hip_bf16.h>
#include <math.h>

#define T_DIM 8000
#define K_DIM 100
#define DEPTH 20

typedef __attribute__((ext_vector_type(2))) float v2f;
typedef __attribute__((ext_vector_type(4))) float v4f;
typedef __attribute__((ext_vector_type(8))) float v8f;

// Fast, numerically stable softplus: max(x,0) + log(1 + exp(-|x|)).
// Uses hardware v_exp_f32 / v_log_f32. When exp(-|x|) underflows the
// log term vanishes and the result is exactly max(x,0).
__device__ __forceinline__ float fast_softplus(float x) {
    float ax = __builtin_fabsf(x);
    float mx = fmaxf(x, 0.0f);
    return mx + __logf(1.0f + __expf(-ax));
}

// ---------------------------------------------------------------------------
// Stage 0a: Alpha = Alpha_full * gsupp ; kern[dd,i] = Beta[i]*exp(-(dd+1)*Beta[i])
//           also zero the loss accumulator (d_out[0]).
// ---------------------------------------------------------------------------
__global__ __launch_bounds__(256) void stage0_prep(
    const float* __restrict__ AlphaFull, const int* __restrict__ gsupp,
    const float* __restrict__ Beta, float* __restrict__ alphaM,
    float* __restrict__ kern, float* __restrict__ lossPtr)
{
    int idx = blockIdx.x * 256 + threadIdx.x;
    if (idx < K_DIM * K_DIM)
        alphaM[idx] = AlphaFull[idx] * (float)gsupp[idx];
    if (idx < DEPTH * K_DIM) {
        int i = idx % K_DIM;
        int dd = idx / K_DIM;
        float b = Beta[i];
        kern[idx] = b * expf(-(float)(dd + 1) * b);
    }
    if (idx == 0) *lossPtr = 0.0f;
}

// ---------------------------------------------------------------------------
// Stage 0b: base[k] = mean_t speeds[t,k] / 3
// ---------------------------------------------------------------------------
__global__ __launch_bounds__(256) void stage0_base(
    const float* __restrict__ speeds, float* __restrict__ base)
{
    __shared__ float red[256];
    int k = blockIdx.x;
    float s = 0.0f;
    for (int t = threadIdx.x; t < T_DIM; t += 256)
        s += speeds[(size_t)t * K_DIM + k];
    red[threadIdx.x] = s;
    __syncthreads();
    for (int off = 128; off > 0; off >>= 1) {
        if (threadIdx.x < off) red[threadIdx.x] += red[threadIdx.x + off];
        __syncthreads();
    }
    if (threadIdx.x == 0)
        base[k] = red[0] / (float)T_DIM / 3.0f;
}

// ---------------------------------------------------------------------------
// Stage 2 (the WMMA kernel): one block per timestep t.
//   E[dd,i]  = kern[dd,i] * speeds[t,i]          (A operand, 20x100 padded to 32)
//   AG[i,j]  = dgraph[t,i,j] * Alpha[i,j]        (B operand, stored transposed)
//   C[dd,t,j] = softplus( E[dd,:] @ AG[:,j] )    via V_WMMA_F32_16X16X4_F32
// dgraph is streamed exactly once -> non-temporal loads so the 320MB stream
// does not evict the L2-resident 64MB Cbuf needed by stage 3.
// ---------------------------------------------------------------------------
__global__ __launch_bounds__(256) void stage2_wmma(
    const float* __restrict__ speeds, const float* __restrict__ dgraph,
    const float* __restrict__ alphaM, const float* __restrict__ kern,
    float* __restrict__ Cbuf)
{
    // AG transposed: sAGT[j][i], j = 0..111 (rows 100..111 zero padding)
    __shared__ __align__(16) float sAGT[112 * K_DIM];
    // E: sE[dd][i], dd = 0..31 (rows 20..31 zero padding)
    __shared__ __align__(16) float sE[32 * K_DIM];

    const int t   = blockIdx.x;
    const int tid = threadIdx.x;

    // zero the padding rows
    for (int idx = tid; idx < 12 * K_DIM; idx += 256) sAGT[100 * K_DIM + idx] = 0.0f;
    for (int idx = tid; idx < 12 * K_DIM; idx += 256) sE[20 * K_DIM + idx]    = 0.0f;

    // load dgraph[t] (contiguous 10000 floats), fuse Alpha mask, store transposed.
    // Rows are 100 floats = 25 float4s, so a 4-vector never crosses a row.
    const v4f* g4 = (const v4f*)(dgraph + (size_t)t * (K_DIM * K_DIM));
    const v4f* a4 = (const v4f*)alphaM;
    for (int idx4 = tid; idx4 < (K_DIM * K_DIM) / 4; idx4 += 256) {
        v4f g = __builtin_nontemporal_load(g4 + idx4);   // TH=NT stream
        v4f a = a4[idx4];                                // L2-hot, RT
        int flat = idx4 * 4;
        int i = flat / K_DIM;
        int j = flat % K_DIM;
        sAGT[(j + 0) * K_DIM + i] = g.x * a.x;
        sAGT[(j + 1) * K_DIM + i] = g.y * a.y;
        sAGT[(j + 2) * K_DIM + i] = g.z * a.z;
        sAGT[(j + 3) * K_DIM + i] = g.w * a.w;
    }

    // E[dd,i] = kern[dd,i] * speeds[t,i]
    const float* srow = speeds + (size_t)t * K_DIM;
    for (int idx = tid; idx < DEPTH * K_DIM; idx += 256) {
        int i = idx % K_DIM;
        sE[idx] = kern[idx] * srow[i];
    }
    __syncthreads();

    const int wave = tid >> 5;   // uniform per wave -> EXEC all-ones inside branch
    const int lane = tid & 31;
    if (wave < 7) {
        const int lh = lane & 15;
        const int hi = lane >> 4;       // 0: K pair (k0,k0+1); 1: (k0+2,k0+3)
        const int j0 = wave * 16;

        v8f acc0 = {};                  // deltas 0..15
        v8f acc1 = {};                  // deltas 16..19 (+ zero-padded rows)

        // A operand (16x4 f32): lanes 0-15 hold M rows, VGPR0/1 = K pair;
        // lanes 16-31 hold same M rows at K+2/K+3. B mirrors with N across lanes.
        const float* eBase0 = sE   + (0  + lh) * K_DIM + 2 * hi;
        const float* eBase1 = sE   + (16 + lh) * K_DIM + 2 * hi;
        const float* bBase  = sAGT + (j0 + lh) * K_DIM + 2 * hi;

        #pragma unroll 5
        for (int k0 = 0; k0 < K_DIM; k0 += 4) {
            v2f a0 = *(const v2f*)(eBase0 + k0);
            v2f a1 = *(const v2f*)(eBase1 + k0);
            v2f b  = *(const v2f*)(bBase  + k0);
            acc0 = __builtin_amdgcn_wmma_f32_16x16x4_f32(
                false, a0, false, b, (short)0, acc0, false, false);
            acc1 = __builtin_amdgcn_wmma_f32_16x16x4_f32(
                false, a1, false, b, (short)0, acc1, false, false);
        }

        // C/D layout: VGPR v, lanes 0-15 -> (M=v, N=lane); lanes 16-31 -> (M=v+8).
        const int j = j0 + lh;
        if (j < K_DIM) {
            #pragma unroll
            for (int v = 0; v < 8; ++v) {
                int dd0 = v + 8 * hi;
                Cbuf[((size_t)dd0 * T_DIM + t) * K_DIM + j] = fast_softplus(acc0[v]);
                int dd1 = 16 + v + 8 * hi;
                if (dd1 < DEPTH)
                    Cbuf[((size_t)dd1 * T_DIM + t) * K_DIM + j] = fast_softplus(acc1[v]);
            }
        }
    }
}

// ---------------------------------------------------------------------------
// Stage 3: pred1[j,t] = sum_{dd < min(D,t)} C[dd, t-dd-1, j]; pred0[j,t] = base[j];
//          loss = mean((speeds[t,j] - pred1 - pred0)^2). C is L2-resident (64MB).
// Output stores are write-once -> non-temporal.
// ---------------------------------------------------------------------------
__global__ __launch_bounds__(256) void stage3_gather(
    const float* __restrict__ speeds, const float* __restrict__ Cbuf,
    const float* __restrict__ base, float* __restrict__ out)
{
    __shared__ float red[256];
    const int j = blockIdx.y;
    const int t = blockIdx.x * 256 + threadIdx.x;

    float sq = 0.0f;
    if (t < T_DIM) {
        float s = 0.0f;
        const int dmax = (t < DEPTH) ? t : DEPTH;
        for (int dd = 0; dd < dmax; ++dd)
            s += Cbuf[((size_t)dd * T_DIM + (t - dd - 1)) * K_DIM + j];
        float b = base[j];
        // out layout: [loss(1)] [pred0 (K,T)] [pred1 (K,T)]
        __builtin_nontemporal_store(b, out + 1 + (size_t)j * T_DIM + t);
        __builtin_nontemporal_store(
            s, out + 1 + (size_t)K_DIM * T_DIM + (size_t)j * T_DIM + t);
        float x = speeds[(size_t)t * K_DIM + j];
        float e = x - s - b;
        sq = e * e;
    }
    red[threadIdx.x] = sq;
    __syncthreads();
    for (int off = 128; off > 0; off >>= 1) {
        if (threadIdx.x < off) red[threadIdx.x] += red[threadIdx.x + off];
        __syncthreads();
    }
    if (threadIdx.x == 0)
        atomicAdd(out, red[0] * (1.0f / (float)((size_t)K_DIM * T_DIM)));
}

// ---------------------------------------------------------------------------
// Launch
// ---------------------------------------------------------------------------
extern "C" void kernel_launch(void* const* d_in, const int* in_sizes, int n_in,
                              void* d_out, int out_size, void* d_ws, size_t ws_size,
                              hipStream_t stream) {
    const float* speeds    = (const float*)d_in[0];   // [T,K]
    const float* dgraph    = (const float*)d_in[1];   // [T,K,K]
    const float* Beta      = (const float*)d_in[2];   // [K]
    const float* AlphaFull = (const float*)d_in[3];   // [K,K]
    const int*   gsupp     = (const int*)d_in[4];     // [K,K]
    // d_in[5] = depth scalar (== DEPTH == 20, fixed by the harness)

    float* out = (float*)d_out;
    float* ws  = (float*)d_ws;
    float* alphaM = ws;              // 10000 floats
    float* kern   = ws + 10240;      // 2000 floats
    float* base   = ws + 12288;      // 100 floats
    float* Cbuf   = ws + 12416;      // DEPTH*T*K = 16M floats (64 MB, L2-resident)

    stage0_prep<<<(K_DIM * K_DIM + 255) / 256, 256, 0, stream>>>(
        AlphaFull, gsupp, Beta, alphaM, kern, out);
    stage0_base<<<K_DIM, 256, 0, stream>>>(speeds, base);
    stage2_wmma<<<T_DIM, 256, 0, stream>>>(speeds, dgraph, alphaM, kern, Cbuf);
    dim3 g3((T_DIM + 255) / 256, K_DIM);
    stage3_gather<<<g3, 256, 0, stream>>>(speeds, Cbuf, base, out);
}